// LSTMEncoder_5076651344431
// MI455X (gfx1250) — compile-verified
//
#include <hip/hip_runtime.h>
#include <cstdint>
#include <cstddef>

#define HDIM 1024
#define TSEQ 4096
#define G4   4096          // 4*HDIM gate rows
#define NBLK 64            // persistent blocks for the recurrence
#define THREADS 256
#define W_STRIDE 1028      // padded LDS row stride in floats (bank-conflict avoidance, 16B aligned)

typedef float v2f __attribute__((ext_vector_type(2)));
typedef float v8f __attribute__((ext_vector_type(8)));

// ---------------- Phase 0: zero h double-buffer + barrier counter ----------------
__global__ void init_ws_kernel(float* __restrict__ h_bufs, unsigned* __restrict__ cnt) {
  int i = blockIdx.x * blockDim.x + threadIdx.x;
  if (i < 2 * HDIM) h_bufs[i] = 0.0f;
  if (i == 0) *cnt = 0u;
}

// ---------------- Phase 1: gx = emb[tokens] @ w_ih^T + (b_ih + b_hh) -------------
// One wave32 per 16x64 output strip (4 N-tiles): A-fragment reused across 4 WMMAs.
__global__ __launch_bounds__(256) void gx_gemm_kernel(
    const int* __restrict__ tokens, const float* __restrict__ emb,
    const float* __restrict__ w_ih, const float* __restrict__ b_ih,
    const float* __restrict__ b_hh, float* __restrict__ gx)
{
  const int wave = threadIdx.x >> 5;
  const int lane = threadIdx.x & 31;
  const int wid  = blockIdx.x * 8 + wave;     // 16384 waves total
  const int tm = wid >> 6;                    // t-tile      (0..255)
  const int tq = wid & 63;                    // 64-col quad (0..63)
  const int half = lane >> 4;                 // lane group (K split)
  const int l16  = lane & 15;

  const int m  = tm * 16 + l16;               // A row (time step)
  const int n0 = tq * 64 + l16;               // first B row / C column of the quad
  const float* __restrict__ arow = emb + (size_t)tokens[m] * HDIM;
  const float* __restrict__ br0 = w_ih + (size_t)(n0)      * HDIM;
  const float* __restrict__ br1 = w_ih + (size_t)(n0 + 16) * HDIM;
  const float* __restrict__ br2 = w_ih + (size_t)(n0 + 32) * HDIM;
  const float* __restrict__ br3 = w_ih + (size_t)(n0 + 48) * HDIM;

  v8f acc0 = {}, acc1 = {}, acc2 = {}, acc3 = {};
  #pragma unroll 4
  for (int k0 = 0; k0 < HDIM; k0 += 4) {
    const int ko = k0 + 2 * half;
    // A 16x4 layout: VGPR0/1 = K {0,1} (lanes 0-15) or K {2,3} (lanes 16-31)
    v2f a  = *(const v2f*)(arow + ko);
    v2f b0 = *(const v2f*)(br0 + ko);
    v2f b1 = *(const v2f*)(br1 + ko);
    v2f b2 = *(const v2f*)(br2 + ko);
    v2f b3 = *(const v2f*)(br3 + ko);
    acc0 = __builtin_amdgcn_wmma_f32_16x16x4_f32(false, a, false, b0, (short)0, acc0, false, false);
    acc1 = __builtin_amdgcn_wmma_f32_16x16x4_f32(false, a, false, b1, (short)0, acc1, false, false);
    acc2 = __builtin_amdgcn_wmma_f32_16x16x4_f32(false, a, false, b2, (short)0, acc2, false, false);
    acc3 = __builtin_amdgcn_wmma_f32_16x16x4_f32(false, a, false, b3, (short)0, acc3, false, false);
  }

  #pragma unroll
  for (int q = 0; q < 4; ++q) {
    const int nq = n0 + q * 16;
    const float bias = b_ih[nq] + b_hh[nq];
    const v8f* accp = (q == 0) ? &acc0 : (q == 1) ? &acc1 : (q == 2) ? &acc2 : &acc3;
    #pragma unroll
    for (int i = 0; i < 8; ++i) {
      // C/D layout: VGPR i holds M = i (lanes 0-15) or M = i+8 (lanes 16-31), N = l16
      int row = tm * 16 + i + 8 * half;
      gx[(size_t)row * G4 + nq] = (*accp)[i] + bias;
    }
  }
}

// ---------------- Grid-wide barrier (monotonic counter) --------------------------
__device__ inline void grid_barrier(unsigned* cnt, unsigned* target_ref) {
  __threadfence();          // make this thread's global writes visible at device scope
  __syncthreads();
  if (threadIdx.x == 0) {
    unsigned tgt = *target_ref + NBLK;
    *target_ref = tgt;
    __hip_atomic_fetch_add(cnt, 1u, __ATOMIC_ACQ_REL, __HIP_MEMORY_SCOPE_AGENT);
    while (__hip_atomic_load(cnt, __ATOMIC_ACQUIRE, __HIP_MEMORY_SCOPE_AGENT) < tgt) {
      __builtin_amdgcn_s_sleep(2);
    }
  }
  __syncthreads();
}

// Async global -> LDS copy of 16 bytes (CDNA5 path, tracked by ASYNCcnt).
// LDS destination = low 32 bits of the generic pointer (wave-relative LDS offset).
__device__ inline void async_copy_b128(void* lds_dst, const void* gsrc) {
  unsigned ldsa = (unsigned)(uintptr_t)lds_dst;
  asm volatile("global_load_async_to_lds_b128 %0, %1, off"
               :: "v"(ldsa), "v"(gsrc) : "memory");
}
__device__ inline void async_wait0() {
  asm volatile("s_wait_asynccnt 0" ::: "memory");
}

// ---------------- Phase 2: persistent LSTM recurrence, w_hh resident in LDS ------
// Block b owns hidden units j in [b*16, b*16+16) -> 64 gate rows, 256KB of w_hh in LDS.
// Thread mapping: localrow = tx&63 (gate*16 + j_local), kpart = tx>>6 (K chunk of 256).
__global__ __launch_bounds__(THREADS) void lstm_recurrence_kernel(
    const float* __restrict__ gx, const float* __restrict__ w_hh,
    float* __restrict__ h_bufs, unsigned* __restrict__ cnt,
    float* __restrict__ out)
{
  extern __shared__ float w_lds[];                  // [64][W_STRIDE]
  __shared__ __align__(16) float h_sm[HDIM];
  __shared__ float red_sm[THREADS];
  __shared__ float gate_sm[64];

  const int tx = threadIdx.x;
  const int b  = blockIdx.x;
  const int localrow = tx & 63;
  const int kpart    = tx >> 6;
  const int gate     = localrow >> 4;
  const int j_local  = localrow & 15;
  const int grow     = gate * HDIM + b * 16 + j_local;   // global gate row

  // Stage this block's 64 rows of w_hh into LDS via async global->LDS b128 copies.
  for (int q = tx; q < 64 * (HDIM / 4); q += THREADS) {
    int lr = q >> 8;                 // 256 float4 per row
    int k4 = q & 255;
    int gg = lr >> 4;
    int jj = lr & 15;
    async_copy_b128(&w_lds[lr * W_STRIDE + k4 * 4],
                    &w_hh[((size_t)(gg * HDIM + b * 16 + jj)) * HDIM + k4 * 4]);
  }
  async_wait0();

  float c_reg  = 0.0f;      // valid in threads tx < 16
  float h_last = 0.0f;
  unsigned target = 0;
  const float* wrow = w_lds + localrow * W_STRIDE + kpart * 256;
  __syncthreads();

  for (int t = 0; t < TSEQ; ++t) {
    // h_{t-1}: 4KB from global double buffer straight into LDS (async, one b128/thread)
    const float* h_src = h_bufs + (t & 1) * HDIM;
    async_copy_b128(&h_sm[tx * 4], h_src + tx * 4);
    async_wait0();
    __syncthreads();

    // partial dot: 256-element K chunk, float4 LDS reads
    const float* hk = h_sm + kpart * 256;
    float acc = 0.0f;
    #pragma unroll 8
    for (int kk = 0; kk < 256; kk += 4) {
      const float4 w4 = *(const float4*)(wrow + kk);
      const float4 h4 = *(const float4*)(hk + kk);
      acc += w4.x * h4.x + w4.y * h4.y + w4.z * h4.z + w4.w * h4.w;
    }
    red_sm[tx] = acc;
    __syncthreads();

    // reduce 4 K-chunks, add gx, apply gate nonlinearity
    if (tx < 64) {
      float v = red_sm[tx] + red_sm[tx + 64] + red_sm[tx + 128] + red_sm[tx + 192]
              + gx[(size_t)t * G4 + grow];
      float a = (gate == 2) ? tanhf(v) : (1.0f / (1.0f + __expf(-v)));
      gate_sm[tx] = a;
    }
    __syncthreads();

    // cell update for the 16 hidden units this block owns
    if (tx < 16) {
      float iv = gate_sm[tx];
      float fv = gate_sm[16 + tx];
      float gv = gate_sm[32 + tx];
      float ov = gate_sm[48 + tx];
      float c = fv * c_reg + iv * gv;
      float h = ov * tanhf(c);
      c_reg = c; h_last = h;
      int j = b * 16 + tx;
      out[(size_t)t * HDIM + j] = h;                   // hs[t]
      h_bufs[((t + 1) & 1) * HDIM + j] = h;            // publish h_t
    }
    grid_barrier(cnt, &target);
  }

  if (tx < 16) {
    int j = b * 16 + tx;
    out[(size_t)TSEQ * HDIM + j]        = h_last;      // h_n
    out[(size_t)TSEQ * HDIM + HDIM + j] = c_reg;       // c_n
  }
}

// ---------------- Host launcher ---------------------------------------------------
extern "C" void kernel_launch(void* const* d_in, const int* in_sizes, int n_in,
                              void* d_out, int out_size, void* d_ws, size_t ws_size,
                              hipStream_t stream) {
  const int*   tokens = (const int*)d_in[0];
  const float* emb    = (const float*)d_in[1];
  const float* w_ih   = (const float*)d_in[2];
  const float* w_hh   = (const float*)d_in[3];
  const float* b_ih   = (const float*)d_in[4];
  const float* b_hh   = (const float*)d_in[5];
  float* out = (float*)d_out;

  char* ws = (char*)d_ws;
  const size_t gx_bytes = (size_t)TSEQ * G4 * sizeof(float);      // 64 MB
  float*    gx     = (float*)ws;
  float*    h_bufs = (float*)(ws + gx_bytes);                      // 2 * HDIM
  unsigned* cnt    = (unsigned*)(ws + gx_bytes + 2 * HDIM * sizeof(float) + 256);

  init_ws_kernel<<<8, 256, 0, stream>>>(h_bufs, cnt);

  // 16384 waves, each a 16x64 strip; 8 waves per block
  gx_gemm_kernel<<<2048, 256, 0, stream>>>(tokens, emb, w_ih, b_ih, b_hh, gx);

  const size_t smem = (size_t)64 * W_STRIDE * sizeof(float);       // 263168 B
  (void)hipFuncSetAttribute(reinterpret_cast<const void*>(lstm_recurrence_kernel),
                            hipFuncAttributeMaxDynamicSharedMemorySize, (int)smem);
  lstm_recurrence_kernel<<<NBLK, THREADS, smem, stream>>>(gx, w_hh, h_bufs, cnt, out);
}